// MambaBlock_7421703488073
// MI455X (gfx1250) — compile-verified
//
#include <hip/hip_runtime.h>
#include <hip/hip_bf16.h>

typedef __attribute__((ext_vector_type(2))) float v2f;
typedef __attribute__((ext_vector_type(8))) float v8f;

#define D_MODEL 512
#define D_STATE 16
#define D_INNER 1024
#define BATCH   2
#define LENGTH  2048
#define MROWS   (BATCH * LENGTH)   // 4096
#define NCHUNK  16
#define TCHUNK  (LENGTH / NCHUNK)  // 128

#define ACT_NONE     0
#define ACT_SILU     1
#define ACT_SOFTPLUS 2

template <int ACT>
__device__ __forceinline__ float apply_act(float v) {
    if constexpr (ACT == ACT_SILU) {
        return v * (1.0f / (1.0f + __expf(-v)));
    } else if constexpr (ACT == ACT_SOFTPLUS) {
        return (v > 20.0f) ? v : log1pf(__expf(v));
    }
    return v;
}

// ---------------------------------------------------------------------------
// fp32 WMMA GEMM:  out[MROWS,N] = act( A'[MROWS,K] @ W[K,N] + bias )
// A' is A (LDA-strided), optionally im2col'd over NTAPS shifted copies
// (SAME conv, pad_left=1): out[t] = sum_tap A[t+tap-1] @ W_tap, rows masked
// at per-batch sequence boundaries.
//
// Block: 256 threads = 8 waves; block tile 64(M) x 128(N).
// Wave w: M sub-tile = (w>>1)*16, N group = (w&1)*64; four 16x16 accumulators
// -> 16 WMMAs per 16-K chunk per wave off a single staged A fragment each kk.
// N/K/LDA are compile-time so all B-fragment loads are base + immediate.
// ---------------------------------------------------------------------------
template <int N, int K, int LDA, int NTAPS, int ACT>
__global__ __launch_bounds__(256)
void gemm_wmma_f32(const float* __restrict__ A,
                   const float* __restrict__ W,      // [NTAPS][K][N] row-major
                   const float* __restrict__ bias,   // [N] or nullptr
                   float* __restrict__ out)
{
    const int tid  = threadIdx.x;
    const int lane = tid & 31;
    const int wave = tid >> 5;
    const int half = lane >> 4;      // K-pair select (A,B) / M+8 select (C)
    const int mr   = lane & 15;      // M index (A) / N index (B,C)
    const int wm   = wave >> 1;      // 0..3  -> M sub-tile
    const int wn   = wave & 1;       // 0..1  -> N 64-col group
    const int tileM   = blockIdx.y * 64;
    const int tileN   = blockIdx.x * 128;
    const int colBase = tileN + wn * 64;

    __shared__ float As[16][65];     // [k][m], padded row stride

    v8f c[4] = {{}, {}, {}, {}};

    // per-lane, loop-invariant part of the B-fragment address
    const int laneoff = 2 * half * N + colBase + mr;

    for (int tap = 0; tap < NTAPS; ++tap) {
        const float* Wt = W + (size_t)tap * K * N;
        const int shift = (NTAPS == 4) ? (tap - 1) : 0;

        for (int k0 = 0; k0 < K; k0 += 16) {
            __syncthreads();
            {   // cooperative stage of A block: 64 rows x 16 k
                const int m   = tid >> 2;
                const int kq  = (tid & 3) * 4;
                const int row = tileM + m;
                bool valid = true;
                if (NTAPS == 4) {
                    const int ts = (row & (LENGTH - 1)) + shift;
                    valid = (ts >= 0) && (ts < LENGTH);
                }
                float v0 = 0.f, v1 = 0.f, v2 = 0.f, v3 = 0.f;
                if (valid) {
                    const float* ap = A + (size_t)(row + shift) * LDA + k0 + kq;
                    v0 = ap[0]; v1 = ap[1]; v2 = ap[2]; v3 = ap[3];
                }
                As[kq + 0][m] = v0;
                As[kq + 1][m] = v1;
                As[kq + 2][m] = v2;
                As[kq + 3][m] = v3;
            }
            __syncthreads();

            const int arow = wm * 16 + mr;
            const float* wk = Wt + (size_t)k0 * N + laneoff;   // one bump per chunk
            #pragma unroll
            for (int kk = 0; kk < 16; kk += 4) {
                v2f a;
                a.x = As[kk + 2 * half + 0][arow];
                a.y = As[kk + 2 * half + 1][arow];
                #pragma unroll
                for (int j = 0; j < 4; ++j) {
                    v2f b;
                    b.x = wk[kk * N + j * 16];        // compile-time immediates
                    b.y = wk[kk * N + N + j * 16];
                    c[j] = __builtin_amdgcn_wmma_f32_16x16x4_f32(
                            false, a, false, b, (short)0, c[j], false, false);
                }
            }
        }
    }

    // epilogue: C/D layout — VGPR r holds M = r (lanes 0-15) / r+8 (lanes 16-31)
    #pragma unroll
    for (int j = 0; j < 4; ++j) {
        #pragma unroll
        for (int r = 0; r < 8; ++r) {
            const int row = tileM + wm * 16 + half * 8 + r;
            const int col = colBase + j * 16 + mr;
            float v = c[j][r];
            if (ACT != ACT_NONE) v += bias[col];
            v = apply_act<ACT>(v);
            out[(size_t)row * N + col] = v;
        }
    }
}

// ---------------------------------------------------------------------------
// B/C projections: Bm[row,s] = xc[row,:] @ w_B[:,s] + b_B[s]; likewise Cm.
// ---------------------------------------------------------------------------
__global__ __launch_bounds__(256)
void bc_proj_kernel(const float* __restrict__ xc,
                    const float* __restrict__ wB, const float* __restrict__ bB,
                    const float* __restrict__ wC, const float* __restrict__ bC,
                    float* __restrict__ Bm, float* __restrict__ Cm)
{
    const int row = blockIdx.x;
    __shared__ float xr[D_INNER];
    __shared__ float partial[8][32];

    for (int i = threadIdx.x; i < D_INNER; i += 256)
        xr[i] = xc[(size_t)row * D_INNER + i];
    __syncthreads();

    const int lane = threadIdx.x & 31;
    const int wave = threadIdx.x >> 5;
    const float* Wsel = (lane < 16) ? wB : wC;
    const int s = lane & 15;

    float acc = 0.f;
    const int kbeg = wave * (D_INNER / 8);
    #pragma unroll 4
    for (int k = kbeg; k < kbeg + (D_INNER / 8); ++k)
        acc += xr[k] * Wsel[(size_t)k * D_STATE + s];
    partial[wave][lane] = acc;
    __syncthreads();

    if (threadIdx.x < 32) {
        float sum = 0.f;
        #pragma unroll
        for (int w = 0; w < 8; ++w) sum += partial[w][threadIdx.x];
        const int s2 = threadIdx.x & 15;
        if (threadIdx.x < 16) Bm[(size_t)row * D_STATE + s2] = sum + bB[s2];
        else                  Cm[(size_t)row * D_STATE + s2] = sum + bC[s2];
    }
}

// ---------------------------------------------------------------------------
// Blocked weighted prefix-sum scan.
// hidden[t,n,s] = sum_{u<=t} 0.95^{u+1} * delta[u,n]*xc[u,n]*B[u,s]
// ---------------------------------------------------------------------------
__global__ __launch_bounds__(256)
void scan_chunk_sums(const float* __restrict__ delta,
                     const float* __restrict__ xc,
                     const float* __restrict__ Bm,
                     float* __restrict__ S)
{
    const int b = blockIdx.z, c = blockIdx.y;
    const int n = blockIdx.x * 256 + threadIdx.x;
    __shared__ float Bs[TCHUNK][D_STATE];

    const int base = b * LENGTH + c * TCHUNK;
    for (int i = threadIdx.x; i < TCHUNK * D_STATE; i += 256)
        (&Bs[0][0])[i] = Bm[(size_t)base * D_STATE + i];
    __syncthreads();

    float acc[D_STATE];
    #pragma unroll
    for (int s = 0; s < D_STATE; ++s) acc[s] = 0.f;

    float dec = __powf(0.95f, (float)(c * TCHUNK + 1));
    for (int t = 0; t < TCHUNK; ++t) {
        const size_t idx = (size_t)(base + t) * D_INNER + n;
        const float w = delta[idx] * xc[idx] * dec;
        #pragma unroll
        for (int s = 0; s < D_STATE; ++s) acc[s] += w * Bs[t][s];
        dec *= 0.95f;
    }
    const size_t so = (((size_t)(b * NCHUNK + c)) * D_INNER + n) * D_STATE;
    #pragma unroll
    for (int s = 0; s < D_STATE; ++s) S[so + s] = acc[s];
}

__global__ __launch_bounds__(256)
void scan_prefix(float* __restrict__ S)
{
    const int idx = blockIdx.x * 256 + threadIdx.x;   // over B*D_INNER*D_STATE
    const int s = idx & 15;
    const int n = (idx >> 4) & (D_INNER - 1);
    const int b = idx / (D_INNER * D_STATE);
    float run = 0.f;
    for (int c = 0; c < NCHUNK; ++c) {
        const size_t o = (((size_t)(b * NCHUNK + c)) * D_INNER + n) * D_STATE + s;
        const float v = S[o];
        S[o] = run;
        run += v;
    }
}

__global__ __launch_bounds__(256)
void scan_final(const float* __restrict__ delta,
                const float* __restrict__ xc,
                const float* __restrict__ Bm,
                const float* __restrict__ Cm,
                const float* __restrict__ xproj,   // z = xproj[..., D_INNER:]
                const float* __restrict__ S,       // exclusive chunk prefixes
                float* __restrict__ y)
{
    const int b = blockIdx.z, c = blockIdx.y;
    const int n = blockIdx.x * 256 + threadIdx.x;
    __shared__ float Bs[TCHUNK][D_STATE];
    __shared__ float Cs[TCHUNK][D_STATE];

    const int base = b * LENGTH + c * TCHUNK;
    for (int i = threadIdx.x; i < TCHUNK * D_STATE; i += 256) {
        (&Bs[0][0])[i] = Bm[(size_t)base * D_STATE + i];
        (&Cs[0][0])[i] = Cm[(size_t)base * D_STATE + i];
    }
    __syncthreads();

    float h[D_STATE];
    const size_t so = (((size_t)(b * NCHUNK + c)) * D_INNER + n) * D_STATE;
    #pragma unroll
    for (int s = 0; s < D_STATE; ++s) h[s] = S[so + s];

    float dec = __powf(0.95f, (float)(c * TCHUNK + 1));
    for (int t = 0; t < TCHUNK; ++t) {
        const size_t idx = (size_t)(base + t) * D_INNER + n;
        const float w = delta[idx] * xc[idx] * dec;
        float acc = 0.f;
        #pragma unroll
        for (int s = 0; s < D_STATE; ++s) {
            h[s] += w * Bs[t][s];
            acc  += h[s] * Cs[t][s];
        }
        const float zv = xproj[(size_t)(base + t) * (2 * D_INNER) + D_INNER + n];
        const float sg = 1.0f / (1.0f + __expf(-zv));
        y[idx] = acc * zv * sg;
        dec *= 0.95f;
    }
}

// ---------------------------------------------------------------------------
extern "C" void kernel_launch(void* const* d_in, const int* in_sizes, int n_in,
                              void* d_out, int out_size, void* d_ws, size_t ws_size,
                              hipStream_t stream)
{
    const float* x       = (const float*)d_in[0];   // (B,L,512)
    const float* w_in    = (const float*)d_in[1];   // (512, 2048)
    const float* conv_k  = (const float*)d_in[2];   // (4, 1024, 1024) [tap][K][N]
    const float* conv_b  = (const float*)d_in[3];   // (1024,)
    const float* w_delta = (const float*)d_in[4];   // (1024, 1024)
    const float* b_delta = (const float*)d_in[5];   // (1024,)
    const float* w_B     = (const float*)d_in[6];   // (1024, 16)
    const float* b_B     = (const float*)d_in[7];   // (16,)
    const float* w_C     = (const float*)d_in[8];   // (1024, 16)
    const float* b_C     = (const float*)d_in[9];   // (16,)
    // d_in[10] = A  (dead parameter in reference)
    const float* w_out   = (const float*)d_in[11];  // (1024, 512)

    float* ws    = (float*)d_ws;
    float* xproj = ws;                      ws += (size_t)MROWS * 2 * D_INNER;
    float* xconv = ws;                      ws += (size_t)MROWS * D_INNER;
    float* delta = ws;                      ws += (size_t)MROWS * D_INNER;
    float* Bm    = ws;                      ws += (size_t)MROWS * D_STATE;
    float* Cm    = ws;                      ws += (size_t)MROWS * D_STATE;
    float* S     = ws;                      ws += (size_t)BATCH * NCHUNK * D_INNER * D_STATE;
    float* ybuf  = ws;                      ws += (size_t)MROWS * D_INNER;

    // 1) x_proj = x @ w_in              (M=4096, N=2048, K=512)
    gemm_wmma_f32<2 * D_INNER, D_MODEL, D_MODEL, 1, ACT_NONE>
        <<<dim3((2 * D_INNER) / 128, MROWS / 64), 256, 0, stream>>>(
            x, w_in, nullptr, xproj);

    // 2) x_conv = silu(SAME-conv(x_z) + conv_b): 4 shifted GEMMs over K=1024
    gemm_wmma_f32<D_INNER, D_INNER, 2 * D_INNER, 4, ACT_SILU>
        <<<dim3(D_INNER / 128, MROWS / 64), 256, 0, stream>>>(
            xproj, conv_k, conv_b, xconv);

    // 3) delta = softplus(x_conv @ w_delta + b_delta)
    gemm_wmma_f32<D_INNER, D_INNER, D_INNER, 1, ACT_SOFTPLUS>
        <<<dim3(D_INNER / 128, MROWS / 64), 256, 0, stream>>>(
            xconv, w_delta, b_delta, delta);

    // 4) Bm, Cm projections (N=16 each)
    bc_proj_kernel<<<MROWS, 256, 0, stream>>>(
        xconv, w_B, b_B, w_C, b_C, Bm, Cm);

    // 5-7) blocked weighted-prefix scan + gate
    scan_chunk_sums<<<dim3(D_INNER / 256, NCHUNK, BATCH), 256, 0, stream>>>(
        delta, xconv, Bm, S);
    scan_prefix<<<(BATCH * D_INNER * D_STATE) / 256, 256, 0, stream>>>(S);
    scan_final<<<dim3(D_INNER / 256, NCHUNK, BATCH), 256, 0, stream>>>(
        delta, xconv, Bm, Cm, xproj, S, ybuf);

    // 8) out = y @ w_out                (M=4096, N=512, K=1024)
    gemm_wmma_f32<D_MODEL, D_INNER, D_INNER, 1, ACT_NONE>
        <<<dim3(D_MODEL / 128, MROWS / 64), 256, 0, stream>>>(
            ybuf, w_out, nullptr, (float*)d_out);
}